// WassersteinLossMicro_73710228735012
// MI455X (gfx1250) — compile-verified
//
#include <hip/hip_runtime.h>
#include <hip/hip_bf16.h>
#include <math.h>

typedef __attribute__((ext_vector_type(2))) float v2f;
typedef __attribute__((ext_vector_type(8))) float v8f;

#define NXE 8192
#define NYE 8192
#define NP  75
#define NFT 8
#define EPS 0.05f
// log2-domain scale: log2(e)/EPS — lets the inner loop use bare v_exp_f32
#define SCALE_L2 28.8539008177792681f
#define LN2F 0.6931471805599453f
#define NITER 200

// NOTE: __builtin_amdgcn_exp2f/__builtin_amdgcn_logf are the HW base-2
// exp/log (v_exp_f32 / v_log_f32) per __clang_hip_math.h on this toolchain.

// ---------------------------------------------------------------------------
// Stage 1: extract feature 3, sort each row of 75 descending (rank scatter).
// One wave per row, 8 rows per 256-thread block.
// ---------------------------------------------------------------------------
__global__ void k_sort(const float* __restrict__ x, const float* __restrict__ y,
                       float* __restrict__ xs) {
  __shared__ float rows[8][80];
  const int wave = threadIdx.x >> 5;
  const int lane = threadIdx.x & 31;
  const int r = blockIdx.x * 8 + wave;  // 0..16383 (x rows then y rows)
  const float* src = (r < NXE) ? (x + (size_t)r * NP * NFT)
                               : (y + (size_t)(r - NXE) * NP * NFT);
  for (int k = lane; k < NP; k += 32) rows[wave][k] = src[k * NFT + 3];
  __syncthreads();
  for (int k = lane; k < NP; k += 32) {
    float v = rows[wave][k];
    int rank = 0;
    for (int j = 0; j < NP; ++j) {
      float u = rows[wave][j];
      rank += (u > v) || (u == v && j < k);  // descending, stable
    }
    xs[(size_t)r * NP + rank] = v;
  }
}

// ---------------------------------------------------------------------------
// Stage 2a: h1 = relu([xs | 0,1] @ W1 + b1).  One row per 256-thread block.
// ---------------------------------------------------------------------------
__global__ void k_mlp1(const float* __restrict__ xs, const float* __restrict__ W1,
                       const float* __restrict__ b1, float* __restrict__ h1) {
  __shared__ float in[80];
  const int r = blockIdx.x;
  const int c = threadIdx.x;  // 0..255
  if (c < NP)            in[c] = xs[(size_t)r * NP + c];
  else if (c == NP)      in[c] = 0.0f;  // label 0
  else if (c == NP + 1)  in[c] = 1.0f;  // label 1
  __syncthreads();
  float acc = b1[c];
  for (int i = 0; i < 77; ++i) acc = fmaf(in[i], W1[i * 256 + c], acc);
  h1[(size_t)r * 256 + c] = fmaxf(acc, 0.0f);
}

// ---------------------------------------------------------------------------
// Stage 2b: h2 = relu(h1@W2+b2); z = h2@W3+b3; s = |z|^2.  64 threads/row.
// ---------------------------------------------------------------------------
__global__ void k_mlp2(const float* __restrict__ h1, const float* __restrict__ W2,
                       const float* __restrict__ b2, const float* __restrict__ W3,
                       const float* __restrict__ b3, float* __restrict__ z,
                       float* __restrict__ s) {
  __shared__ float h2s[64];
  __shared__ float zz[2];
  const int r = blockIdx.x;
  const int c = threadIdx.x;  // 0..63
  const float* hrow = h1 + (size_t)r * 256;
  float acc = b2[c];
  for (int i = 0; i < 256; ++i) acc = fmaf(hrow[i], W2[i * 64 + c], acc);
  h2s[c] = fmaxf(acc, 0.0f);
  __syncthreads();
  if (c < 2) {
    float a = b3[c];
    for (int i = 0; i < 64; ++i) a = fmaf(h2s[i], W3[i * 2 + c], a);
    zz[c] = a;
    z[(size_t)r * 2 + c] = a;
  }
  __syncthreads();
  if (c == 0) s[r] = zz[0] * zz[0] + zz[1] * zz[1];
}

// ---------------------------------------------------------------------------
// Stage 3 prep: sum of x_weights; init f=g=0, loga, logb, zero output.
// ---------------------------------------------------------------------------
__global__ void k_wsum(const float* __restrict__ w, float* __restrict__ wsum) {
  __shared__ float red[256];
  const int t = threadIdx.x;
  float a = 0.f;
  for (int i = t; i < NXE; i += 256) a += w[i];
  red[t] = a;
  __syncthreads();
  for (int o = 128; o > 0; o >>= 1) {
    if (t < o) red[t] += red[t + o];
    __syncthreads();
  }
  if (t == 0) wsum[0] = red[0];
}

__global__ void k_init(const float* __restrict__ w, const float* __restrict__ wsum,
                       float* __restrict__ f, float* __restrict__ g,
                       float* __restrict__ loga, float* __restrict__ logb,
                       float* __restrict__ out) {
  const int i = blockIdx.x * blockDim.x + threadIdx.x;
  if (i < NXE) {
    f[i] = 0.f;
    g[i] = 0.f;
    loga[i] = LN2F * (__builtin_amdgcn_logf(w[i]) - __builtin_amdgcn_logf(wsum[0]));
    logb[i] = -LN2F * __builtin_amdgcn_logf((float)NYE);
  }
  if (i == 0) out[0] = 0.f;
}

// ---------------------------------------------------------------------------
// Stage 3: one Sinkhorn half-step (updates the "row" side's dual).
//   vOut_m = EPS * (logw_m - logsumexp_n((vCol_n - M_mn)/EPS))
//   M_mn = max(sRow_m + sCol_n - 2 zRow_m . zCol_n, 0)
// One wave (32 threads) per 16-row strip; 16-col tiles via
// V_WMMA_F32_16X16X4_F32 with C preloaded = sRow_m + sCol_n, A = -2 zRow,
// so the WMMA output IS the raw-M tile.  Two passes (row-max, then
// sum-exp) halve the transcendental count vs online softmax; logsumexp
// runs in the log2 domain so exp is a bare v_exp_f32; the clamp is
// folded as t = min(tb, fma(d,-S,tb)) which avoids canonicalize maxes.
// B is loaded unmasked: lanes 16-31 feed K=2,3 rows which multiply
// A's zeroed K=2,3 entries (finite garbage * 0 == 0).
// ---------------------------------------------------------------------------
__global__ void __launch_bounds__(32)
k_sink(const float* __restrict__ zRow, const float* __restrict__ sRow,
       const float* __restrict__ zCol, const float* __restrict__ sCol,
       const float* __restrict__ vCol, const float* __restrict__ logw,
       float* __restrict__ vOut, int nCol) {
  const int lane = threadIdx.x;
  const int lo = lane & 15;
  const int hi = lane >> 4;  // 0 | 1: selects rows m0+v vs m0+8+v in C/D layout
  const int m0 = blockIdx.x * 16;

  // A matrix 16x4 (K=2 used, K=2..3 zero): lanes 0-15 hold K=0,1; 16-31 hold K=2,3.
  v2f a;
  {
    float z0 = zRow[(m0 + lo) * 2 + 0];
    float z1 = zRow[(m0 + lo) * 2 + 1];
    a.x = hi ? 0.f : -2.f * z0;
    a.y = hi ? 0.f : -2.f * z1;
  }
  float rs[8];
#pragma unroll
  for (int v = 0; v < 8; ++v) rs[v] = sRow[m0 + hi * 8 + v];

  float rmax[8];
#pragma unroll
  for (int v = 0; v < 8; ++v) rmax[v] = -3.0e38f;

  // ---- pass 1: per-row max of t2 = (vCol - M) * log2(e)/EPS ----
  for (int n0 = 0; n0 < nCol; n0 += 16) {
    const int n = n0 + lo;
    v2f b;
    b.x = zCol[n * 2 + 0];
    b.y = zCol[n * 2 + 1];
    float sc = sCol[n];
    float tb = vCol[n] * SCALE_L2;
    v8f c;
#pragma unroll
    for (int v = 0; v < 8; ++v) c[v] = rs[v] + sc;
    v8f d = __builtin_amdgcn_wmma_f32_16x16x4_f32(false, a, false, b, (short)0,
                                                  c, false, false);
#pragma unroll
    for (int v = 0; v < 8; ++v) {
      // t = tb - S*max(d,0) == min(tb, tb - S*d)
      float t = fminf(tb, fmaf(d[v], -SCALE_L2, tb));
      rmax[v] = fmaxf(rmax[v], t);
    }
  }
#pragma unroll
  for (int off = 1; off < 16; off <<= 1) {
#pragma unroll
    for (int v = 0; v < 8; ++v)
      rmax[v] = fmaxf(rmax[v], __shfl_xor(rmax[v], off, 32));
  }

  // ---- pass 2: sum 2^(t2 - rowmax2) ----
  float rsum[8];
#pragma unroll
  for (int v = 0; v < 8; ++v) rsum[v] = 0.f;

  for (int n0 = 0; n0 < nCol; n0 += 16) {
    const int n = n0 + lo;
    v2f b;
    b.x = zCol[n * 2 + 0];
    b.y = zCol[n * 2 + 1];
    float sc = sCol[n];
    float tb = vCol[n] * SCALE_L2;
    v8f c;
#pragma unroll
    for (int v = 0; v < 8; ++v) c[v] = rs[v] + sc;
    v8f d = __builtin_amdgcn_wmma_f32_16x16x4_f32(false, a, false, b, (short)0,
                                                  c, false, false);
#pragma unroll
    for (int v = 0; v < 8; ++v) {
      float t = fminf(tb, fmaf(d[v], -SCALE_L2, tb));
      rsum[v] += __builtin_amdgcn_exp2f(t - rmax[v]);
    }
  }
#pragma unroll
  for (int off = 1; off < 16; off <<= 1) {
#pragma unroll
    for (int v = 0; v < 8; ++v) rsum[v] += __shfl_xor(rsum[v], off, 32);
  }

  if (lo < 8) {
    const int v = lo;
    const int row = m0 + hi * 8 + v;
    // back to natural log
    float lse = LN2F * (rmax[v] + __builtin_amdgcn_logf(rsum[v]));
    vOut[row] = EPS * (logw[row] - lse);
  }
}

// ---------------------------------------------------------------------------
// Stage 4: cost = sum_ij exp((f_i + g_j - M)/EPS) * M.  Same WMMA tiling,
// exp in log2 domain.
// ---------------------------------------------------------------------------
__global__ void __launch_bounds__(32)
k_cost(const float* __restrict__ zx, const float* __restrict__ sx,
       const float* __restrict__ zy, const float* __restrict__ sy,
       const float* __restrict__ f, const float* __restrict__ g,
       float* __restrict__ out, int nCol) {
  const int lane = threadIdx.x;
  const int lo = lane & 15;
  const int hi = lane >> 4;
  const int m0 = blockIdx.x * 16;

  v2f a;
  {
    float z0 = zx[(m0 + lo) * 2 + 0];
    float z1 = zx[(m0 + lo) * 2 + 1];
    a.x = hi ? 0.f : -2.f * z0;
    a.y = hi ? 0.f : -2.f * z1;
  }
  float rs[8], fs2[8];
#pragma unroll
  for (int v = 0; v < 8; ++v) {
    rs[v] = sx[m0 + hi * 8 + v];
    fs2[v] = f[m0 + hi * 8 + v] * SCALE_L2;  // hoisted f_i * log2(e)/EPS
  }
  float acc = 0.f;
  for (int n0 = 0; n0 < nCol; n0 += 16) {
    const int n = n0 + lo;
    v2f b;
    b.x = zy[n * 2 + 0];
    b.y = zy[n * 2 + 1];
    float sc = sy[n];
    float g2 = g[n] * SCALE_L2;
    v8f c;
#pragma unroll
    for (int v = 0; v < 8; ++v) c[v] = rs[v] + sc;
    v8f d = __builtin_amdgcn_wmma_f32_16x16x4_f32(false, a, false, b, (short)0,
                                                  c, false, false);
#pragma unroll
    for (int v = 0; v < 8; ++v) {
      float M = fmaxf(d[v], 0.f);
      float arg = fmaf(M, -SCALE_L2, fs2[v] + g2);
      acc = fmaf(__builtin_amdgcn_exp2f(arg), M, acc);
    }
  }
#pragma unroll
  for (int off = 1; off < 32; off <<= 1) acc += __shfl_xor(acc, off, 32);
  if (lane == 0) atomicAdd(out, acc);
}

// ---------------------------------------------------------------------------
extern "C" void kernel_launch(void* const* d_in, const int* in_sizes, int n_in,
                              void* d_out, int out_size, void* d_ws, size_t ws_size,
                              hipStream_t stream) {
  (void)in_sizes; (void)n_in; (void)out_size; (void)ws_size;
  const float* x  = (const float*)d_in[0];
  const float* y  = (const float*)d_in[1];
  const float* xw = (const float*)d_in[2];
  const float* W1 = (const float*)d_in[3];
  const float* b1 = (const float*)d_in[4];
  const float* W2 = (const float*)d_in[5];
  const float* b2 = (const float*)d_in[6];
  const float* W3 = (const float*)d_in[7];
  const float* b3 = (const float*)d_in[8];
  float* out = (float*)d_out;

  float* ws = (float*)d_ws;
  size_t off = 0;
  float* xs   = ws + off; off += (size_t)16384 * NP;    // sorted rows (x then y)
  float* h1   = ws + off; off += (size_t)16384 * 256;   // hidden layer 1
  float* z    = ws + off; off += (size_t)16384 * 2;     // latents (zx then zy)
  float* s    = ws + off; off += 16384;                 // |z|^2
  float* fv   = ws + off; off += NXE;
  float* gv   = ws + off; off += NYE;
  float* loga = ws + off; off += NXE;
  float* logb = ws + off; off += NYE;
  float* wsum = ws + off; off += 1;

  k_sort<<<(16384 / 8), 256, 0, stream>>>(x, y, xs);
  k_mlp1<<<16384, 256, 0, stream>>>(xs, W1, b1, h1);
  k_mlp2<<<16384, 64, 0, stream>>>(h1, W2, b2, W3, b3, z, s);
  k_wsum<<<1, 256, 0, stream>>>(xw, wsum);
  k_init<<<NXE / 256, 256, 0, stream>>>(xw, wsum, fv, gv, loga, logb, out);

  const float* zx = z;
  const float* zy = z + (size_t)NXE * 2;
  const float* sx = s;
  const float* sy = s + NXE;

  for (int it = 0; it < NITER; ++it) {
    k_sink<<<NXE / 16, 32, 0, stream>>>(zx, sx, zy, sy, gv, loga, fv, NYE);
    k_sink<<<NYE / 16, 32, 0, stream>>>(zy, sy, zx, sx, fv, logb, gv, NXE);
  }
  k_cost<<<NXE / 16, 32, 0, stream>>>(zx, sx, zy, sy, fv, gv, out, NYE);
}